// RandomShiftsAug_2241972928537
// MI455X (gfx1250) — compile-verified
//
#include <hip/hip_runtime.h>
#include <hip/hip_bf16.h>

#define PADW 4
#define H    96
#define CH   9
#define HP   104               // H + 2*PADW
#define ROWF (H * CH)          // 864 floats per image row

// Force a wave-uniform value into an SGPR for inline-asm "s" constraints.
__device__ __forceinline__ unsigned sgpr(unsigned v) {
    return (unsigned)__builtin_amdgcn_readfirstlane((int)v);
}

// Replicates: arange = linspace(-1+eps, 1-eps, HP)[:H]; g = arange[i] + s*2/HP;
// pix = ((g+1)*HP - 1) * 0.5   (mathematically == i + s, but keep fp-faithful)
__device__ __forceinline__ float pixel_coord(int i, float s) {
    const float eps   = 1.0f / (float)HP;
    const float delta = (2.0f - 2.0f * eps) / (float)(HP - 1);
    float a = -1.0f + eps + (float)i * delta;
    float g = a + s * (2.0f / (float)HP);
    return ((g + 1.0f) * (float)HP - 1.0f) * 0.5f;
}

__global__ __launch_bounds__(256) void RandomShiftsAug_tdm_kernel(
    const float* __restrict__ x,      // [N, H, H, CH]
    const int*   __restrict__ shift,  // [N, 2]
    float*       __restrict__ out)    // [N, H, H, CH]
{
    __shared__ float lds[2 * ROWF];   // two staged source rows (6912 B)

    const int y = blockIdx.x;         // output row 0..H-1
    const int n = blockIdx.y;         // image index

    const int sx = shift[2 * n + 0];
    const int sy = shift[2 * n + 1];

    // ---- Row (y) coordinate: uniform across the whole block ----
    float iy  = pixel_coord(y, (float)sy);
    float y0f = floorf(iy);
    int   y0i = (int)y0f;
    float wy1 = iy - y0f;
    float wy0 = 1.0f - wy1;
    // zeros-padding validity on the padded image [0, HP-1]
    float wy0e = (y0i     >= 0 && y0i     <= HP - 1) ? wy0 : 0.0f;
    float wy1e = (y0i + 1 >= 0 && y0i + 1 <= HP - 1) ? wy1 : 0.0f;
    // padded-row -> original-row (edge padding == clamp)
    int yc0 = min(max(y0i,     0), HP - 1);
    int yc1 = min(max(y0i + 1, 0), HP - 1);
    int rA  = min(max(yc0 - PADW, 0), H - 1);
    int rB  = min(max(yc1 - PADW, 0), H - 1);

    // ---- TDM: DMA the two source rows into LDS as one 864x2 tile.
    // Edge clamping folded into the dim0 (row) stride: 0 duplicates row rA.
    if (threadIdx.x < 32) {            // exactly wave 0 issues the DMA
        unsigned ldsAddr = (unsigned)(size_t)(&lds[0]);   // low 32 bits = LDS offset
        unsigned long long ga =
            (unsigned long long)(size_t)(x + ((size_t)n * H + rA) * ROWF);
        unsigned strideElems = (unsigned)(rB - rA) * (unsigned)ROWF;  // 0 or 864

        // D# group 0 (4 dwords) — all forced into SGPRs via readfirstlane
        unsigned d0 = sgpr(1u);                                  // count=1 (valid D#)
        unsigned d1 = sgpr(ldsAddr);                             // lds_addr (bytes)
        unsigned d2 = sgpr((unsigned)(ga & 0xFFFFFFFFu));        // global_addr[31:0]
        unsigned d3 = sgpr((unsigned)((ga >> 32) & 0x01FFFFFFu)  // global_addr[56:32]
                           | (2u << 30));                        // type = 2 ("image")
        // D# group 1 (8 dwords)
        unsigned e0 = sgpr(2u << 16);                            // data_size = 4 bytes
        unsigned e1 = sgpr(((unsigned)ROWF) << 16);              // tensor_dim0 = 864
        unsigned e2 = sgpr(2u << 16);                            // tensor_dim1 = 2
        unsigned e3 = sgpr(((unsigned)ROWF) << 16);              // tile_dim0 = 864
        unsigned e4 = sgpr(2u);                                  // tile_dim1=2, dim2=0
        unsigned e5 = sgpr(strideElems);                         // tensor_dim0_stride
        unsigned e6 = sgpr(0u);                                  // stride hi parts
        unsigned e7 = sgpr(0u);

        // Stage descriptor into fixed, aligned SGPR groups:
        //   group0 -> s[84:87] (4-aligned), group1 -> s[88:95] (8-aligned)
        asm volatile(
            "s_mov_b32 s84, %0\n\t"
            "s_mov_b32 s85, %1\n\t"
            "s_mov_b32 s86, %2\n\t"
            "s_mov_b32 s87, %3\n\t"
            "s_mov_b32 s88, %4\n\t"
            "s_mov_b32 s89, %5\n\t"
            "s_mov_b32 s90, %6\n\t"
            "s_mov_b32 s91, %7\n\t"
            "s_mov_b32 s92, %8\n\t"
            "s_mov_b32 s93, %9\n\t"
            "s_mov_b32 s94, %10\n\t"
            "s_mov_b32 s95, %11\n\t"
            "tensor_load_to_lds s[84:87], s[88:95]\n\t"
            "s_wait_tensorcnt 0x0"
            :
            : "s"(d0), "s"(d1), "s"(d2), "s"(d3),
              "s"(e0), "s"(e1), "s"(e2), "s"(e3),
              "s"(e4), "s"(e5), "s"(e6), "s"(e7)
            : "s84","s85","s86","s87","s88","s89","s90","s91",
              "s92","s93","s94","s95","memory");
    }
    __syncthreads();                   // LDS tile visible to all 8 waves

    // ---- Compute: 864 outputs per block as 216 float4 stores ----
    const int t4 = threadIdx.x;
    if (t4 < ROWF / 4) {
        float4 o;
        float* op = &o.x;
#pragma unroll
        for (int k = 0; k < 4; ++k) {
            int t  = 4 * t4 + k;
            int xi = t / CH;
            int c  = t - xi * CH;

            float ix  = pixel_coord(xi, (float)sx);
            float x0f = floorf(ix);
            int   x0i = (int)x0f;
            float wx1 = ix - x0f;
            float wx0 = 1.0f - wx1;
            float wx0e = (x0i     >= 0 && x0i     <= HP - 1) ? wx0 : 0.0f;
            float wx1e = (x0i + 1 >= 0 && x0i + 1 <= HP - 1) ? wx1 : 0.0f;
            int cA = min(max(min(max(x0i,     0), HP - 1) - PADW, 0), H - 1);
            int cB = min(max(min(max(x0i + 1, 0), HP - 1) - PADW, 0), H - 1);

            float a0 = lds[cA * CH + c];
            float a1 = lds[cB * CH + c];
            float b0 = lds[ROWF + cA * CH + c];
            float b1 = lds[ROWF + cB * CH + c];

            op[k] = wy0e * (wx0e * a0 + wx1e * a1)
                  + wy1e * (wx0e * b0 + wx1e * b1);
        }
        float4* orow = (float4*)(out + ((size_t)n * H + y) * ROWF);
        orow[t4] = o;   // 16B-aligned: row base is 3456B-aligned, t4*16 offsets
    }
}

extern "C" void kernel_launch(void* const* d_in, const int* in_sizes, int n_in,
                              void* d_out, int out_size, void* d_ws, size_t ws_size,
                              hipStream_t stream) {
    const float* x     = (const float*)d_in[0];
    const int*   shift = (const int*)d_in[1];
    float*       out   = (float*)d_out;

    const int N = in_sizes[1] / 2;     // 256 images
    dim3 grid(H, N);
    dim3 block(256);
    RandomShiftsAug_tdm_kernel<<<grid, block, 0, stream>>>(x, shift, out);
}